// GCNLayer_34531537059966
// MI455X (gfx1250) — compile-verified
//
#include <hip/hip_runtime.h>
#include <hip/hip_bf16.h>

// ---------------------------------------------------------------------------
// GCN layer on gfx1250 (MI455X):
//   deg = segment_sum(edge_val, edge_row)
//   dis = rsqrt(deg) (0 for deg==0)
//   agg[r] += dis[r]*ev*dis[c] * features[c]      (wave-per-edge f32 atomics)
//   out = agg @ W^T                               (V_WMMA_F32_16X16X4_F32)
// ---------------------------------------------------------------------------

typedef __attribute__((ext_vector_type(2))) float v2f;
typedef __attribute__((ext_vector_type(8))) float v8f;

#define D_FEAT 32

// ---- zero workspace (agg + deg contiguous) --------------------------------
__global__ void gcn_zero_kernel(float* __restrict__ p, size_t n) {
    size_t i = (size_t)blockIdx.x * blockDim.x + threadIdx.x;
    if (i < n) p[i] = 0.0f;
}

// ---- degree: deg[row[e]] += val[e] ----------------------------------------
__global__ void gcn_degree_kernel(const int* __restrict__ er,
                                  const float* __restrict__ ev,
                                  float* __restrict__ deg, int E) {
    int e = blockIdx.x * blockDim.x + threadIdx.x;
    if (e < E) atomicAdd(&deg[er[e]], ev[e]);
}

// ---- deg -> d^{-1/2} in place ---------------------------------------------
__global__ void gcn_rsqrt_kernel(float* __restrict__ deg, int N) {
    int i = blockIdx.x * blockDim.x + threadIdx.x;
    if (i < N) {
        float d = deg[i];
        deg[i] = (d > 0.0f) ? rsqrtf(d) : 0.0f;
    }
}

// ---- SpMM: one wave per edge, one feature dim per lane --------------------
// 128B coalesced gather of features[c][:] + 32-lane coalesced f32 atomic add
// into agg[r][:]. Everything is L2-resident (192MB L2 >> 13MB arrays).
__global__ void gcn_spmm_kernel(const int* __restrict__ er,
                                const int* __restrict__ ec,
                                const float* __restrict__ ev,
                                const float* __restrict__ dis,
                                const float* __restrict__ feat,
                                float* __restrict__ agg, int E) {
    int wave = (int)(((size_t)blockIdx.x * blockDim.x + threadIdx.x) >> 5);
    if (wave >= E) return;
    int lane = threadIdx.x & 31;
    int r = er[wave];
    int c = ec[wave];
    float v = dis[r] * ev[wave] * dis[c];
    float f = feat[(size_t)c * D_FEAT + lane];
    atomicAdd(&agg[(size_t)r * D_FEAT + lane], v * f);
}

// ---- GEMM via V_WMMA_F32_16X16X4_F32: out = agg @ W^T ---------------------
// One wave computes a 16(node) x 32(out-feature) tile as two 16x16 WMMA
// accumulators, chaining 8 K=4 steps (D_IN = 32). Exact fp32 semantics.
//
// A (16x4 f32) per-lane layout: lanes 0-15 hold M=lane, K={k, k+1};
//                               lanes 16-31 hold M=lane-16, K={k+2, k+3}.
// B (4x16 f32) mirrors A: lane supplies B[K={kk,kk+1}][N=lane&15].
// C/D (16x16 f32): VGPR i holds row M=i (lanes 0-15) / M=i+8 (lanes 16-31).
__global__ void gcn_gemm_wmma_kernel(const float* __restrict__ agg,
                                     const float* __restrict__ W,   // [32][32] = [out][in]
                                     float* __restrict__ out, int nTiles) {
    int wave = (int)(((size_t)blockIdx.x * blockDim.x + threadIdx.x) >> 5);
    if (wave >= nTiles) return;      // wave-uniform exit keeps EXEC all-ones
    int lane = threadIdx.x & 31;
    int half = lane >> 4;            // 0: K pair {k,k+1}, 1: {k+2,k+3}
    int l15  = lane & 15;

    const float* aRow = agg + (size_t)(wave * 16 + l15) * D_FEAT;

    v8f acc0 = {};
    v8f acc1 = {};
#pragma unroll
    for (int k = 0; k < D_FEAT; k += 4) {
        int kk = k + half * 2;
        v2f a, b0, b1;
        a.x  = aRow[kk];
        a.y  = aRow[kk + 1];
        // B[k][n] = W^T[k][n] = W[n][k]; n-tile 0: n = l15, tile 1: n = 16+l15
        b0.x = W[l15 * D_FEAT + kk];
        b0.y = W[l15 * D_FEAT + kk + 1];
        b1.x = W[(16 + l15) * D_FEAT + kk];
        b1.y = W[(16 + l15) * D_FEAT + kk + 1];
        acc0 = __builtin_amdgcn_wmma_f32_16x16x4_f32(
            false, a, false, b0, (short)0, acc0, false, false);
        acc1 = __builtin_amdgcn_wmma_f32_16x16x4_f32(
            false, a, false, b1, (short)0, acc1, false, false);
    }

    int mBase = wave * 16 + half * 8;
#pragma unroll
    for (int i = 0; i < 8; ++i) {
        out[(size_t)(mBase + i) * D_FEAT + l15]      = acc0[i];
        out[(size_t)(mBase + i) * D_FEAT + 16 + l15] = acc1[i];
    }
}

// ---- scalar tail for node rows beyond the last full 16-tile ----------------
__global__ void gcn_gemm_tail_kernel(const float* __restrict__ agg,
                                     const float* __restrict__ W,
                                     float* __restrict__ out,
                                     int rowStart, int N) {
    int idx = blockIdx.x * blockDim.x + threadIdx.x;
    int row = rowStart + (idx >> 5);
    int o = idx & 31;
    if (row >= N) return;
    float s = 0.0f;
#pragma unroll
    for (int k = 0; k < D_FEAT; ++k)
        s += agg[(size_t)row * D_FEAT + k] * W[o * D_FEAT + k];
    out[(size_t)row * D_FEAT + o] = s;
}

extern "C" void kernel_launch(void* const* d_in, const int* in_sizes, int n_in,
                              void* d_out, int out_size, void* d_ws, size_t ws_size,
                              hipStream_t stream) {
    const int*   er   = (const int*)d_in[0];
    const int*   ec   = (const int*)d_in[1];
    const float* ev   = (const float*)d_in[2];
    const float* feat = (const float*)d_in[3];
    const float* W    = (const float*)d_in[4];
    float* out = (float*)d_out;

    int E = in_sizes[0];
    int N = in_sizes[3] / D_FEAT;

    float* agg = (float*)d_ws;                    // N*32 floats
    float* deg = agg + (size_t)N * D_FEAT;        // N floats (becomes d^{-1/2})

    // 1) zero agg+deg (contiguous)
    size_t zn = (size_t)N * (D_FEAT + 1);
    gcn_zero_kernel<<<(unsigned)((zn + 255) / 256), 256, 0, stream>>>(agg, zn);

    // 2) degrees
    gcn_degree_kernel<<<(E + 255) / 256, 256, 0, stream>>>(er, ev, deg, E);

    // 3) d^{-1/2}
    gcn_rsqrt_kernel<<<(N + 255) / 256, 256, 0, stream>>>(deg, N);

    // 4) SpMM scatter-add: one wave per edge
    size_t spmmThreads = (size_t)E * 32;
    gcn_spmm_kernel<<<(unsigned)((spmmThreads + 255) / 256), 256, 0, stream>>>(
        er, ec, ev, deg, feat, agg, E);

    // 5) dense GEMM out = agg @ W^T via fp32 WMMA (one wave per 16-node tile)
    int nTiles = N / 16;
    if (nTiles > 0) {
        size_t gemmThreads = (size_t)nTiles * 32;
        gcn_gemm_wmma_kernel<<<(unsigned)((gemmThreads + 255) / 256), 256, 0, stream>>>(
            agg, W, out, nTiles);
    }
    int rowStart = nTiles * 16;
    if (rowStart < N) {
        int tailThreads = (N - rowStart) * 32;
        gcn_gemm_tail_kernel<<<(tailThreads + 255) / 256, 256, 0, stream>>>(
            agg, W, out, rowStart, N);
    }
}